// AttentionNet_20761871909584
// MI455X (gfx1250) — compile-verified
//
#include <hip/hip_runtime.h>

typedef __attribute__((ext_vector_type(2))) float v2f;
typedef __attribute__((ext_vector_type(8))) float v8f;

#define BH 32
#define LL 128
#define DD 128
#define LDSP 132   // padded LDS row stride (floats): (m*132+d)%64 banks distinct per m

// ---------------------------------------------------------------------------
// Kernel A: Xp = X @ W1^T + b1  and  Yp = Y @ W2^T + b2, via v_wmma_f32_16x16x4_f32.
// One wave per 16x16 output tile. tileId: [sel(2)][bh(32)][trow(8)][tcol(8)] = 4096 tiles.
// A fragment (16x4 f32): lane m = lane&15 holds A[m][k0+kSub], A[m][k0+kSub+1]
//   with kSub = 2*(lane>>4)  -> contiguous float2 from the X/Y row.
// B fragment (4x16 f32): lane n = lane&15 holds B[k0+kSub][n], B[k0+kSub+1][n]
//   where B[k][n] = W[n][k]  -> contiguous float2 from W's row n (row-major (D,D)).
// ---------------------------------------------------------------------------
__global__ __launch_bounds__(256) void proj_gemm_wmma(
    const float* __restrict__ X,  const float* __restrict__ Y,
    const float* __restrict__ W1, const float* __restrict__ b1,
    const float* __restrict__ W2, const float* __restrict__ b2,
    float* __restrict__ Xp, float* __restrict__ Yp)
{
    const int wave   = threadIdx.x >> 5;
    const int lane   = threadIdx.x & 31;
    const int tileId = blockIdx.x * 8 + wave;      // 0..4095
    const int sel    = tileId >> 11;               // 0 -> X path, 1 -> Y path
    const int rem    = tileId & 2047;
    const int bh     = rem >> 6;
    const int t      = rem & 63;
    const int trow   = t >> 3;
    const int tcol   = t & 7;

    const float* In   = sel ? Y  : X;
    const float* W    = sel ? W2 : W1;
    const float* bias = sel ? b2 : b1;
    float*       Out  = sel ? Yp : Xp;

    const int m    = lane & 15;
    const int kSub = (lane >> 4) << 1;             // 0 or 2
    const float* arow = In + (size_t)(bh * LL + trow * 16 + m) * DD + kSub;
    const float* brow = W  + (size_t)(tcol * 16 + m) * DD + kSub;

    v8f c = {};
#pragma unroll
    for (int k0 = 0; k0 < DD; k0 += 4) {
        v2f a = *(const v2f*)(arow + k0);
        v2f b = *(const v2f*)(brow + k0);
        // (neg_a, A, neg_b, B, c_mod, C, reuse_a, reuse_b)
        c = __builtin_amdgcn_wmma_f32_16x16x4_f32(false, a, false, b,
                                                  (short)0, c, false, false);
    }

    const float bv = bias[tcol * 16 + m];
    const int rbase = trow * 16 + ((lane >> 4) << 3);   // C/D layout: lanes>=16 hold M+8
    float* ocol = Out + (size_t)(bh * LL) * DD + tcol * 16 + m;
#pragma unroll
    for (int i = 0; i < 8; ++i)
        ocol[(size_t)(rbase + i) * DD] = c[i] + bv;
}

// ---------------------------------------------------------------------------
// Kernel B: out[bh,l,m] = b3 + sum_d tanh(Xp[bh,l,d]*Yp[bh,m,d]) * w3[d]
// Block = one (bh, 16-l-tile, 16-m-tile). LDS-staged tiles, padded stride.
// tanh via the gfx1250 single-instruction transcendental (v_tanh_f32).
// ---------------------------------------------------------------------------
__device__ __forceinline__ float fast_tanh(float x) {
#if __has_builtin(__builtin_amdgcn_tanhf)
    return __builtin_amdgcn_tanhf(x);
#else
    float r;
    asm("v_tanh_f32 %0, %1" : "=v"(r) : "v"(x));
    return r;
#endif
}

__global__ __launch_bounds__(256) void pair_tanh_reduce(
    const float* __restrict__ Xp, const float* __restrict__ Yp,
    const float* __restrict__ w3, const float* __restrict__ b3p,
    float* __restrict__ out)
{
    __shared__ float Xs[16 * LDSP];
    __shared__ float Ys[16 * LDSP];
    __shared__ float w3s[DD];

    const int tid = threadIdx.x;
    const int blk = blockIdx.x;          // 0..2047
    const int bh  = blk >> 6;
    const int t   = blk & 63;
    const int lt  = t >> 3;
    const int mt  = t & 7;

    const float* xb = Xp + (size_t)(bh * LL + lt * 16) * DD;
    const float* yb = Yp + (size_t)(bh * LL + mt * 16) * DD;
    for (int idx = tid; idx < 16 * DD; idx += 256) {
        const int r = idx >> 7;          // row in tile
        const int d = idx & 127;         // column
        Xs[r * LDSP + d] = xb[(size_t)r * DD + d];
        Ys[r * LDSP + d] = yb[(size_t)r * DD + d];
    }
    if (tid < DD) w3s[tid] = w3[tid];
    __syncthreads();

    const int ml = tid & 15;
    const int ll = tid >> 4;
    const float* xr = Xs + ll * LDSP;    // 132*4 bytes = multiple of 16 -> b128-aligned
    const float* yr = Ys + ml * LDSP;

    float acc = 0.f;
#pragma unroll
    for (int d0 = 0; d0 < DD; d0 += 4) {
        const float4 xv = *(const float4*)(xr + d0);
        const float4 yv = *(const float4*)(yr + d0);
        acc += fast_tanh(xv.x * yv.x) * w3s[d0 + 0];
        acc += fast_tanh(xv.y * yv.y) * w3s[d0 + 1];
        acc += fast_tanh(xv.z * yv.z) * w3s[d0 + 2];
        acc += fast_tanh(xv.w * yv.w) * w3s[d0 + 3];
    }

    out[(size_t)(bh * LL + lt * 16 + ll) * LL + mt * 16 + ml] = acc + b3p[0];
}

// ---------------------------------------------------------------------------
extern "C" void kernel_launch(void* const* d_in, const int* in_sizes, int n_in,
                              void* d_out, int out_size, void* d_ws, size_t ws_size,
                              hipStream_t stream) {
    (void)in_sizes; (void)n_in; (void)out_size; (void)ws_size;
    const float* X  = (const float*)d_in[0];
    const float* Y  = (const float*)d_in[1];
    const float* W1 = (const float*)d_in[2];
    const float* b1 = (const float*)d_in[3];
    const float* W2 = (const float*)d_in[4];
    const float* b2 = (const float*)d_in[5];
    const float* w3 = (const float*)d_in[6];
    const float* b3 = (const float*)d_in[7];
    float* out = (float*)d_out;

    float* Xp = (float*)d_ws;                      // BH*L*D floats = 2 MB
    float* Yp = Xp + (size_t)BH * LL * DD;         // next 2 MB

    // 4096 wave-tiles / 8 waves per block = 512 blocks
    proj_gemm_wmma<<<512, 256, 0, stream>>>(X, Y, W1, b1, W2, b2, Xp, Yp);
    // 32 bh * 8 * 8 tiles = 2048 blocks
    pair_tanh_reduce<<<2048, 256, 0, stream>>>(Xp, Yp, w3, b3, out);
}